// VoxelShuffle_40948218200800
// MI455X (gfx1250) — compile-verified
//
#include <hip/hip_runtime.h>
#include <hip/hip_bf16.h>
#include <cstdint>

// -------------------------------------------------------------------------
// VoxelShuffle (R=2, V=8, C=64, sequential pairing)
//   out[0] : feats       = features reshaped  -> byte-identical 256MB copy
//   out[1] : new_indices = per input row (b,z,y,x), 8 rows (b,2z+i,2y+j,2x+k)
//            with v = i*4 + k*2 + j  =>  i=v>>2, k=(v>>1)&1, j=v&1
// Pure data movement: ~656 MB total traffic, ~28us at 23.3 TB/s (HBM-bound,
// zero FLOPs -> WMMA not applicable). Bulk copy uses the CDNA5 async LDS
// path (global_load_async_to_lds_b128 / global_store_async_from_lds_b128,
// ASYNCcnt), keeping all bulk data out of VGPRs.
// -------------------------------------------------------------------------

#if defined(__HIP_DEVICE_COMPILE__) && \
    __has_builtin(__builtin_amdgcn_global_load_async_to_lds_b128) && \
    __has_builtin(__builtin_amdgcn_global_store_async_from_lds_b128)
#define VS_USE_ASYNC 1
#else
#define VS_USE_ASYNC 0
#endif

#if VS_USE_ASYNC
typedef int v4i __attribute__((ext_vector_type(4)));
typedef __attribute__((address_space(1))) void vs_gvoid;  // global void
typedef __attribute__((address_space(3))) void vs_lvoid;  // LDS void
typedef __attribute__((address_space(1))) v4i  vs_gv4i;   // global int4-vec
typedef __attribute__((address_space(3))) v4i  vs_lv4i;   // LDS int4-vec

// generic -> AS void* -> same-AS typed pointer (both casts probe-verified)
__device__ __forceinline__ vs_gv4i* vs_as_global(const void* p) {
  return (vs_gv4i*)((vs_gvoid*)p);
}
__device__ __forceinline__ vs_lv4i* vs_as_lds(void* p) {
  return (vs_lv4i*)((vs_lvoid*)p);
}
#endif

__device__ __forceinline__ void vs_wait_async0() {
#if defined(__HIP_DEVICE_COMPILE__)
#if __has_builtin(__builtin_amdgcn_s_wait_asynccnt)
  __builtin_amdgcn_s_wait_asynccnt(0);
#else
  asm volatile("s_wait_asynccnt 0" ::: "memory");
#endif
#endif
}

// ---- Kernel 1: bulk feats copy via async HBM -> LDS -> HBM (B128) -------
// Guard-free: host launches exactly n4 / (VS_THREADS*VS_ELEMS) full blocks,
// so every lane is in-bounds and no exec-mask predication is generated.
// Each thread's LDS slot is private to its own wave, so the only sync is
// the per-wave s_wait_asynccnt 0 between the load burst and store burst.
// Trailing async stores are drained by S_ENDPGM's implicit wait-idle.
#define VS_THREADS 256
#define VS_ELEMS   4
#define VS_TILE    (VS_THREADS * VS_ELEMS)   // float4s per block

__global__ void __launch_bounds__(VS_THREADS)
vs_feats_copy_kernel(const float4* __restrict__ src,
                     float4* __restrict__ dst) {
  __shared__ float4 smem[VS_TILE];
  const long long base =
      (long long)blockIdx.x * VS_TILE + threadIdx.x;

#if VS_USE_ASYNC
  #pragma unroll
  for (int e = 0; e < VS_ELEMS; ++e) {
    __builtin_amdgcn_global_load_async_to_lds_b128(
        vs_as_global(src + base + (long long)e * VS_THREADS),
        vs_as_lds(&smem[threadIdx.x + e * VS_THREADS]),
        /*offset=*/0, /*cpol=*/0);
  }
  vs_wait_async0();  // loads done (only loads outstanding on ASYNCcnt)
  #pragma unroll
  for (int e = 0; e < VS_ELEMS; ++e) {
    __builtin_amdgcn_global_store_async_from_lds_b128(
        vs_as_global(dst + base + (long long)e * VS_THREADS),
        vs_as_lds(&smem[threadIdx.x + e * VS_THREADS]),
        /*offset=*/0, /*cpol=*/0);
  }
#else
  #pragma unroll
  for (int e = 0; e < VS_ELEMS; ++e) {
    const long long idx = base + (long long)e * VS_THREADS;
    dst[idx] = src[idx];
  }
#endif
}

// Tail kernel for non-multiple sizes (not taken for N=1M, C=64).
__global__ void __launch_bounds__(256)
vs_feats_tail_kernel(const float4* __restrict__ src,
                     float4* __restrict__ dst,
                     long long start, long long n4) {
  const long long idx = start + (long long)blockIdx.x * blockDim.x + threadIdx.x;
  if (idx < n4) dst[idx] = src[idx];
}

// ---- Kernel 2: index expansion (one thread per output row) --------------
// tid -> n = tid>>3, v = tid&7. Lanes 0..7 share one 16B input row
// (single line fetch after coalescing); each lane writes one int4,
// fully coalesced 512B per wave per store instruction.
__global__ void __launch_bounds__(256)
vs_expand_indices_kernel(const int4* __restrict__ idx_in,
                         int4* __restrict__ idx_out,
                         int n_out_rows) {
  const int t = blockIdx.x * blockDim.x + threadIdx.x;
  if (t >= n_out_rows) return;
  const int n = t >> 3;
  const int v = t & 7;
  const int4 r = idx_in[n];          // (b, z, y, x)
  const int i = (v >> 2) & 1;        // added to z
  const int k = (v >> 1) & 1;        // added to x
  const int j = v & 1;               // added to y
  int4 o;
  o.x = r.x;
  o.y = (r.y << 1) + i;
  o.z = (r.z << 1) + j;
  o.w = (r.w << 1) + k;
  idx_out[t] = o;
}

extern "C" void kernel_launch(void* const* d_in, const int* in_sizes, int n_in,
                              void* d_out, int out_size, void* d_ws, size_t ws_size,
                              hipStream_t stream) {
  (void)n_in; (void)out_size; (void)d_ws; (void)ws_size;

  const float* features = (const float*)d_in[0];   // [N, 64] fp32
  const int*   indices  = (const int*)d_in[1];     // [N, 4]  int32

  float* out = (float*)d_out;
  const long long n_feat = (long long)in_sizes[0];       // N*C floats
  const long long n_rows = (long long)in_sizes[1] / 4;   // N
  const long long n4     = n_feat / 4;                   // float4 count

  // feats copy: guard-free async B128 staged copy over full tiles
  const long long full_blocks = n4 / VS_TILE;            // 16384 for this size
  if (full_blocks > 0) {
    vs_feats_copy_kernel<<<(int)full_blocks, VS_THREADS, 0, stream>>>(
        (const float4*)features, (float4*)out);
  }
  const long long rem_start = full_blocks * VS_TILE;
  const long long rem = n4 - rem_start;                  // 0 for this size
  if (rem > 0) {
    const int tblocks = (int)((rem + 255) / 256);
    vs_feats_tail_kernel<<<tblocks, 256, 0, stream>>>(
        (const float4*)features, (float4*)out, rem_start, n4);
  }

  // new_indices: written as raw int32 bits after the feats region
  const long long n_out_rows = n_rows * 8;
  int4* idx_out = (int4*)(out + n_feat);
  const int blocks2 = (int)((n_out_rows + 255) / 256);
  vs_expand_indices_kernel<<<blocks2, 256, 0, stream>>>(
      (const int4*)indices, idx_out, (int)n_out_rows);
}